// SoftNMS_81965155877223
// MI455X (gfx1250) — compile-verified
//
#include <hip/hip_runtime.h>
#include <stdint.h>

#define NBOX   8192
#define NWG    32
#define NTHR   256
#define KBLK   32
#define NBLK   (NBOX / KBLK)   /* 256 pivot blocks */
#define THRESH 0.001f

// d_ws layout (bytes):
//   [0    , 1028)  barrier counters u32[257]   (memset to 0 each launch)
//   [2048 , 3072)  per-block active masks u32[256]
//   [4096 , 36864) sorted scores f32[8192]     (written before read)

__global__ __launch_bounds__(NTHR) void softnms_gfx1250(
    const float* __restrict__ boxes_in,
    const float* __restrict__ scores_in,
    float* __restrict__ out,          // [0,32768): sorted boxes, [32768,40960): scores
    unsigned* __restrict__ ws) {

  // 64 KB LDS: bitonic sort keys; later reused as the 32-entry pivot-box buffer.
  __shared__ unsigned long long skey[NBOX];
  float4* lbuf = reinterpret_cast<float4*>(skey);

  const unsigned tid  = threadIdx.x;
  const unsigned wg   = blockIdx.x;
  const unsigned lane = tid & 31u;

  unsigned* bar    = ws;                                   // u32[257]
  unsigned* flags  = ws + 512;                             // byte 2048
  float*    ssc    = reinterpret_cast<float*>(ws + 1024);  // byte 4096
  float*    oboxes = out;
  float*    oscore = out + 4 * NBOX;

  // ---------------- Phase A: WG0 stable-sorts (descending score) in LDS ----
  if (wg == 0) {
    for (unsigned i = tid; i < NBOX; i += NTHR) {
      unsigned f = __float_as_uint(scores_in[i]);
      unsigned m = (f & 0x80000000u) ? ~f : (f | 0x80000000u); // monotone ↑ map
      // ascending sort of (~m, i)  ==  descending by score, ties by index ↑
      skey[i] = ((unsigned long long)(~m) << 32) | (unsigned long long)i;
    }
    __syncthreads();
    for (unsigned k = 2; k <= NBOX; k <<= 1) {
      for (unsigned j = k >> 1; j > 0; j >>= 1) {
        for (unsigned t = tid; t < NBOX / 2; t += NTHR) {
          unsigned lo = 2u * t - (t & (j - 1u));
          unsigned hi = lo + j;
          bool asc = ((lo & k) == 0u);
          unsigned long long a = skey[lo], c = skey[hi];
          if ((a > c) == asc) { skey[lo] = c; skey[hi] = a; }
        }
        __syncthreads();
      }
    }
    const float4* b4 = reinterpret_cast<const float4*>(boxes_in);
    float4*       o4 = reinterpret_cast<float4*>(oboxes);
    for (unsigned i = tid; i < NBOX; i += NTHR) {
      unsigned idx = (unsigned)skey[i];
      o4[i]  = b4[idx];
      ssc[i] = scores_in[idx];
    }
    __threadfence();
  }

  // ---------------- global barrier (slot 0): sorted data published ---------
  __syncthreads();
  if (tid == 0) {
    __hip_atomic_fetch_add(&bar[0], 1u, __ATOMIC_RELEASE, __HIP_MEMORY_SCOPE_AGENT);
    while (__hip_atomic_load(&bar[0], __ATOMIC_ACQUIRE, __HIP_MEMORY_SCOPE_AGENT) < NWG)
      __builtin_amdgcn_s_sleep(1);
  }
  __syncthreads();

  // ---------------- Phase B: one element per thread, register resident -----
  const unsigned jg = wg * NTHR + tid;      // element index 0..8191
  float4 B    = reinterpret_cast<const float4*>(oboxes)[jg];
  float  s    = ssc[jg];
  float  areaJ = (B.z - B.x) * (B.w - B.y);
  const unsigned long long gbase = (unsigned long long)oboxes;

  for (unsigned b = 0; b < NBLK; ++b) {
    // --- A: diagonal block, exact sequential semantics inside the owner wave
    if ((jg >> 5) == b) {
      unsigned mask = 0u;
      for (unsigned p = 0; p < KBLK; ++p) {
        float sp = __shfl(s, (int)p);
        if (sp != 0.0f) {                       // uniform across the wave
          mask |= (1u << p);
          float px1 = __shfl(B.x, (int)p);
          float py1 = __shfl(B.y, (int)p);
          float px2 = __shfl(B.z, (int)p);
          float py2 = __shfl(B.w, (int)p);
          float aP  = __shfl(areaJ, (int)p);
          if (lane > p) {
            float w = fminf(px2, B.z) - fmaxf(px1, B.x);
            float h = fminf(py2, B.w) - fmaxf(py1, B.y);
            w = fmaxf(w, 0.0f); h = fmaxf(h, 0.0f);
            float inter = w * h;
            if (inter > 0.0f) {                 // inter==0 -> decay==1 exactly
              float iou = inter / (aP + areaJ - inter);
              s *= (1.0f - iou);                // mirrors reference rounding
            }
          }
          if (s < THRESH) s = 0.0f;             // per-step threshold (all lanes)
        }
      }
      if (lane == 0)
        __hip_atomic_store(&flags[b], mask, __ATOMIC_RELEASE, __HIP_MEMORY_SCOPE_AGENT);
    }
    __syncthreads();

    // --- C: wave0 async-copies this block's 32 pivot boxes into LDS
    //        (gfx1250 global_load_async_to_lds_b128, GVS mode); it drains
    //        under the global-barrier latency. Also prefetch block b+1
    //        (global_prefetch_b8) so the next async copy hits in cache.
    if (tid < 32u) {
      unsigned ldsOff = (unsigned)(unsigned long long)(&lbuf[lane]); // low 32b = LDS offset
      unsigned voff   = (b * KBLK + lane) * 16u;
      asm volatile("global_load_async_to_lds_b128 %0, %1, %2"
                   :: "v"(ldsOff), "v"(voff), "s"(gbase) : "memory");
      if (b + 1u < NBLK)
        __builtin_prefetch(oboxes + ((b + 1u) * KBLK + lane) * 4u, 0, 3);
    }
    // --- D: inter-workgroup barrier (one-shot slot 1+b)
    if (tid == 0) {
      __hip_atomic_fetch_add(&bar[1 + b], 1u, __ATOMIC_RELEASE, __HIP_MEMORY_SCOPE_AGENT);
      while (__hip_atomic_load(&bar[1 + b], __ATOMIC_ACQUIRE, __HIP_MEMORY_SCOPE_AGENT) < NWG)
        __builtin_amdgcn_s_sleep(1);
    }
    // --- E: wave0 drains its async copy, then publish LDS to the workgroup
    if (tid < 32u) asm volatile("s_wait_asynccnt 0x0" ::: "memory");
    __syncthreads();

    // --- G: trailing update for all elements past this block
    unsigned mask = __hip_atomic_load(&flags[b], __ATOMIC_RELAXED, __HIP_MEMORY_SCOPE_AGENT);
    if (mask != 0u && jg >= (b + 1u) * KBLK) {
      for (unsigned p = 0; p < KBLK; ++p) {
        if (mask & (1u << p)) {
          float4 P = lbuf[p];                   // LDS broadcast read
          float w = fminf(P.z, B.z) - fmaxf(P.x, B.x);
          float h = fminf(P.w, B.w) - fmaxf(P.y, B.y);
          w = fmaxf(w, 0.0f); h = fmaxf(h, 0.0f);
          float inter = w * h;
          if (inter > 0.0f) {                   // ~98% of pairs skip the divide
            float aP  = (P.z - P.x) * (P.w - P.y);
            float iou = inter / (aP + areaJ - inter);
            s *= (1.0f - iou);
          }
        }
      }
      if (s < THRESH) s = 0.0f;                 // equivalent to per-step threshold
    }
  }

  oscore[jg] = s;
}

extern "C" void kernel_launch(void* const* d_in, const int* in_sizes, int n_in,
                              void* d_out, int out_size, void* d_ws, size_t ws_size,
                              hipStream_t stream) {
  const float* boxes  = (const float*)d_in[0];   // (8192,4) f32
  const float* scores = (const float*)d_in[1];   // (8192,)  f32
  float* out = (float*)d_out;                    // 40960 f32: boxes then scores
  unsigned* ws = (unsigned*)d_ws;

  // zero the one-shot barrier counters + flags every launch (graph-replay safe)
  hipMemsetAsync(d_ws, 0, 4096, stream);
  hipLaunchKernelGGL(softnms_gfx1250, dim3(NWG), dim3(NTHR), 0, stream,
                     boxes, scores, out, ws);
}